// QueryGenerator_54546084659715
// MI455X (gfx1250) — compile-verified
//
#include <hip/hip_runtime.h>
#include <hip/hip_bf16.h>

// ---------------- problem constants (match reference) ----------------
#define BB   2
#define HH   512
#define WW   512
#define CIN  32
#define CH   64
#define QD   32
#define OUTC 68          // 2*QD + 4
#define NCLS 8
#define THR  0.95f

#if defined(__gfx1250__)
#define QG_ASYNC_LDS 1
#endif

typedef __attribute__((ext_vector_type(2))) float v2f;
typedef __attribute__((ext_vector_type(8))) float v8f;

// ---------------- small RNG helpers (deterministic, counter-based) ----------------
__device__ __forceinline__ unsigned pcg_hash(unsigned x) {
    x = x * 747796405u + 2891336453u;
    x = ((x >> ((x >> 28) + 4u)) ^ x) * 277803737u;
    return (x >> 22) ^ x;
}
__device__ __forceinline__ float u01(unsigned h) {
    return ((float)(h >> 8) + 0.5f) * (1.0f / 16777216.0f);
}
__device__ __forceinline__ float rnormal(unsigned s) {
    unsigned h1 = pcg_hash(s);
    unsigned h2 = pcg_hash(h1 ^ 0x9E3779B9u);
    float u1 = u01(h1), u2 = u01(h2);
    return sqrtf(-2.0f * __logf(u1)) * __cosf(6.28318530718f * u2);
}
__device__ float rgamma(float a, unsigned s) {
    float boost = 1.0f;
    if (a < 1.0f) {
        float u = u01(pcg_hash(s ^ 0xA511E9B3u));
        boost = __powf(u, 1.0f / fmaxf(a, 1e-6f));
        a += 1.0f;
    }
    float d = a - (1.0f / 3.0f);
    float c = rsqrtf(9.0f * d);
    float vkeep = 1.0f;
    for (int it = 0; it < 16; ++it) {
        float x = rnormal(s + 0x632BE5ABu * (unsigned)(it + 1));
        float v = 1.0f + c * x;
        if (v <= 0.0f) continue;
        v = v * v * v;
        float u = u01(pcg_hash(s + 0x85EBCA6Bu * (unsigned)(it + 1)));
        vkeep = v;
        if (__logf(u) < 0.5f * x * x + d - d * v + d * __logf(v)) break;
    }
    return boost * d * vkeep;
}
__device__ __forceinline__ float rbeta(float a, float b, unsigned s) {
    float ga = rgamma(a, s);
    float gb = rgamma(b, s ^ 0x5BD1E995u);
    return ga / (ga + gb + 1e-20f);
}

// LDS byte offset (wave-relative) from a generic pointer to __shared__ memory:
// per the aperture rules (ISA 10.2), the low 32 bits of a generic LDS address
// are the wave-relative LDS offset, which is what async-to-LDS VDST expects.
__device__ __forceinline__ unsigned qg_lds_off(const void* p) {
    return (unsigned)(size_t)p;
}

// ---------------- kernel 0: zero the dense grid ----------------
__global__ void qg_zero_kernel(float4* p, int n4) {
    int i = blockIdx.x * blockDim.x + threadIdx.x;
    if (i < n4) p[i] = make_float4(0.f, 0.f, 0.f, 0.f);
}

// ---------------- kernel 1: scatter pixel_values into dense ----------------
__global__ void qg_scatter_kernel(const int* __restrict__ idx,
                                  const float* __restrict__ vals,
                                  float* __restrict__ dense, int N) {
    int t = blockIdx.x * blockDim.x + threadIdx.x;
    if (t >= N * CIN) return;
    int n = t >> 5;            // CIN == 32
    int c = t & 31;
    int b = idx[n * 3 + 0], y = idx[n * 3 + 1], x = idx[n * 3 + 2];
    dense[((((size_t)b * HH + y) * WW + x) * CIN) + c] = vals[t];
}

// ---------------- kernel 2: softmax-cdf counts ----------------
__global__ void qg_counts_kernel(const float* __restrict__ logits,
                                 int* __restrict__ counts, int N) {
    int n = blockIdx.x * blockDim.x + threadIdx.x;
    if (n >= N) return;
    float l[NCLS];
    float mx = -1e30f;
    #pragma unroll
    for (int c = 0; c < NCLS; ++c) { l[c] = logits[n * NCLS + c]; mx = fmaxf(mx, l[c]); }
    float s = 0.f;
    #pragma unroll
    for (int c = 0; c < NCLS; ++c) { l[c] = __expf(l[c] - mx); s += l[c]; }
    float inv = 1.0f / s, cdf = 0.f;
    int cnt = 0;
    #pragma unroll
    for (int c = 0; c < NCLS; ++c) {
        cdf += l[c] * inv;
        if (cdf > THR) { cnt = c; break; }
    }
    counts[n] = cnt;
}

// ---------------- kernel 3: single-block exclusive prefix scan ----------------
__global__ void qg_scan_kernel(const int* __restrict__ counts,
                               int* __restrict__ off, int N) {
    __shared__ int buf[256];
    __shared__ int carry;
    int tid = threadIdx.x;
    if (tid == 0) carry = 0;
    __syncthreads();
    for (int base = 0; base < N; base += 256) {
        int i = base + tid;
        int v = (i < N) ? counts[i] : 0;
        buf[tid] = v;
        __syncthreads();
        for (int s = 1; s < 256; s <<= 1) {
            int t = (tid >= s) ? buf[tid - s] : 0;
            __syncthreads();
            buf[tid] += t;
            __syncthreads();
        }
        if (i < N) off[i] = carry + buf[tid] - v;   // exclusive
        __syncthreads();
        if (tid == 0) carry += buf[255];
        __syncthreads();
    }
    if (tid == 0) off[N] = carry;                   // total
}

// ---------------- kernel 4a: default-fill qidx/rowpix with last pixel ----------------
__global__ void qg_fill_kernel(const int* __restrict__ idx,
                               int* __restrict__ qidx_out,
                               int* __restrict__ rowpix, int N, int T) {
    int j = blockIdx.x * blockDim.x + threadIdx.x;
    if (j >= T) return;
    int n = N - 1;
    qidx_out[3 * j + 0] = idx[3 * n + 0];
    qidx_out[3 * j + 1] = idx[3 * n + 1];
    qidx_out[3 * j + 2] = idx[3 * n + 2];
    rowpix[j] = n;
}

// ---------------- kernel 4b: expand repeats ----------------
__global__ void qg_expand_kernel(const int* __restrict__ idx,
                                 const int* __restrict__ off,
                                 const int* __restrict__ counts,
                                 int* __restrict__ qidx_out,
                                 int* __restrict__ rowpix, int N, int T) {
    int n = blockIdx.x * blockDim.x + threadIdx.x;
    if (n >= N) return;
    int b = idx[3 * n + 0], y = idx[3 * n + 1], x = idx[3 * n + 2];
    int start = off[n], cnt = counts[n];
    for (int k = 0; k < cnt; ++k) {
        int j = start + k;
        if (j >= T) break;
        qidx_out[3 * j + 0] = b;
        qidx_out[3 * j + 1] = y;
        qidx_out[3 * j + 2] = x;
        rowpix[j] = n;
    }
}

// ---------------- kernel 5: fused sparse conv3x3 + ReLU + conv1x1 + exp via WMMA ----------------
// Each wave32 owns 16 pixels; each block = 4 waves = 64 pixels.
// Per 3x3 tap: stage im2col slice (64px x 32ch) and w1 tap slice (32x64) into LDS with
// GLOBAL_LOAD_ASYNC_TO_LDS_B128 (ASYNCcnt + s_wait_asynccnt), then V_WMMA_F32_16X16X4_F32.
// GEMM2: h (16x64) x w2 (64x80 zero-padded), 16 K-steps * 5 N-tiles.
#define CONV_WAVES 4
#define CONV_PIX   (CONV_WAVES * 16)
#define SA_STRIDE  36   // floats; 16B-aligned rows, conflict-free A-frag reads
#define SB_STRIDE  68   // floats; 16B-aligned rows

__global__ __launch_bounds__(CONV_WAVES * 32)
void qg_conv_wmma_kernel(const int* __restrict__ idx,
                         const float* __restrict__ w1,   // (3,3,32,64) HWIO flat
                         const float* __restrict__ b1,   // (64)
                         const float* __restrict__ w2,   // (64,68) flat
                         const float* __restrict__ b2,   // (68)
                         const float* __restrict__ dense,
                         float* __restrict__ P,          // (N,68) exp(vae) per pixel
                         int N) {
    __shared__ alignas(16) float sA[CONV_PIX][SA_STRIDE]; // per-tap patch slice
    __shared__ alignas(16) float sB[CIN][SB_STRIDE];      // per-tap w1 slice
    __shared__ float sW2[CH][80];                         // w2 zero-padded to 80 cols
    __shared__ float sH[CONV_WAVES][16][CH + 1];
    __shared__ float sb1[CH];
    __shared__ float sb2[80];
    __shared__ int   sPix[CONV_PIX][3];

    const int tid  = threadIdx.x;
    const int wave = tid >> 5;
    const int lane = tid & 31;
    const int blockPix0 = blockIdx.x * CONV_PIX;

    // stage w2 (padded), biases, pixel coords
    for (int i = tid; i < CH * 80; i += CONV_WAVES * 32) {
        int k = i / 80, n = i % 80;
        sW2[k][n] = (n < OUTC) ? w2[k * OUTC + n] : 0.0f;
    }
    if (tid < CH) sb1[tid] = b1[tid];
    if (tid < 80) sb2[tid] = (tid < OUTC) ? b2[tid] : 0.0f;
    for (int i = tid; i < CONV_PIX; i += CONV_WAVES * 32) {
        int g  = blockPix0 + i;
        int gc = (g < N) ? g : (N - 1);
        sPix[i][0] = idx[gc * 3 + 0];
        sPix[i][1] = idx[gc * 3 + 1];
        sPix[i][2] = idx[gc * 3 + 2];
    }
    __syncthreads();

    const int m     = lane & 15;               // pixel-in-tile (A rows) / col-in-tile (B,C)
    const int khalf = (lane >> 4) << 1;        // 0 for lanes 0-15, 2 for lanes 16-31
    const int moff  = (lane >> 4) * 8;         // C/D row offset

    v8f acc[4] = {};                           // h tile accumulators (16 x 64)

    for (int tap = 0; tap < 9; ++tap) {
        const int dy = tap / 3 - 1;
        const int dx = tap % 3 - 1;
        __syncthreads();   // protect sA/sB from previous tap's readers

        // stage w1 tap slice (32x64 floats) as 16B chunks: 512 chunks / 128 threads
        for (int i = tid; i < (CIN * CH) / 4; i += CONV_WAVES * 32) {
            int k = i >> 4, q = i & 15;
            float* ldst = &sB[k][q * 4];
            const float* g = w1 + ((size_t)tap * CIN + k) * CH + q * 4;
#if QG_ASYNC_LDS
            asm volatile("global_load_async_to_lds_b128 %0, %1, off"
                         :: "v"(qg_lds_off(ldst)), "v"(g) : "memory");
#else
            *(float4*)ldst = *(const float4*)g;
#endif
        }
        // stage patch slice: 64 pixels x 32 ch as 16B chunks (8 per pixel)
        for (int i = tid; i < CONV_PIX * (CIN / 4); i += CONV_WAVES * 32) {
            int p = i >> 3, q = i & 7;
            int bb = sPix[p][0];
            int yy = sPix[p][1] + dy;
            int xx = sPix[p][2] + dx;
            float* ldst = &sA[p][q * 4];
            if ((unsigned)yy < (unsigned)HH && (unsigned)xx < (unsigned)WW) {
                const float* g = dense + ((((size_t)bb * HH + yy) * WW + xx) * CIN) + q * 4;
#if QG_ASYNC_LDS
                asm volatile("global_load_async_to_lds_b128 %0, %1, off"
                             :: "v"(qg_lds_off(ldst)), "v"(g) : "memory");
#else
                *(float4*)ldst = *(const float4*)g;
#endif
            } else {
                *(float4*)ldst = make_float4(0.f, 0.f, 0.f, 0.f);   // zero padding at borders
            }
        }
#if QG_ASYNC_LDS
        asm volatile("s_wait_asynccnt 0x0" ::: "memory");
#endif
        __syncthreads();

        #pragma unroll
        for (int kk = 0; kk < CIN / 4; ++kk) {
            v2f a;
            a.x = sA[wave * 16 + m][kk * 4 + khalf];
            a.y = sA[wave * 16 + m][kk * 4 + khalf + 1];
            #pragma unroll
            for (int nt = 0; nt < 4; ++nt) {
                v2f b;
                b.x = sB[kk * 4 + khalf][nt * 16 + m];
                b.y = sB[kk * 4 + khalf + 1][nt * 16 + m];
                acc[nt] = __builtin_amdgcn_wmma_f32_16x16x4_f32(
                    false, a, false, b, (short)0, acc[nt], false, false);
            }
        }
    }

    // bias + ReLU, stash h tile to LDS for layout change (C-layout -> A-layout)
    #pragma unroll
    for (int nt = 0; nt < 4; ++nt) {
        float bias = sb1[nt * 16 + m];
        #pragma unroll
        for (int v = 0; v < 8; ++v) {
            float h = acc[nt][v] + bias;
            sH[wave][v + moff][nt * 16 + m] = (h > 0.0f) ? h : 0.0f;
        }
    }
    __syncthreads();

    // GEMM2: (16x64) x (64x80)
    v8f acc2[5] = {};
    #pragma unroll
    for (int kk = 0; kk < CH / 4; ++kk) {
        v2f a;
        a.x = sH[wave][m][kk * 4 + khalf];
        a.y = sH[wave][m][kk * 4 + khalf + 1];
        #pragma unroll
        for (int nt = 0; nt < 5; ++nt) {
            v2f b;
            b.x = sW2[kk * 4 + khalf][nt * 16 + m];
            b.y = sW2[kk * 4 + khalf + 1][nt * 16 + m];
            acc2[nt] = __builtin_amdgcn_wmma_f32_16x16x4_f32(
                false, a, false, b, (short)0, acc2[nt], false, false);
        }
    }

    // p = exp(vae + b2), store per-pixel rows
    #pragma unroll
    for (int nt = 0; nt < 5; ++nt) {
        int col = nt * 16 + m;
        if (col < OUTC) {
            float bias = sb2[col];
            #pragma unroll
            for (int v = 0; v < 8; ++v) {
                int p = blockPix0 + wave * 16 + v + moff;
                if (p < N) P[(size_t)p * OUTC + col] = __expf(acc2[nt][v] + bias);
            }
        }
    }
}

// ---------------- kernel 6: queries = mu + exp(logvar) * eps ----------------
__global__ void qg_queries_kernel(const float* __restrict__ P,
                                  const int* __restrict__ rowpix,
                                  float* __restrict__ qout, int T) {
    int t = blockIdx.x * blockDim.x + threadIdx.x;
    if (t >= T * QD) return;
    int j = t >> 5;                // QD == 32
    int d = t & 31;
    int n = rowpix[j];
    float mu = P[(size_t)n * OUTC + d];
    float lv = P[(size_t)n * OUTC + QD + d];
    float eps = rnormal(0xB5297A4Du ^ (unsigned)t);
    qout[t] = mu + __expf(lv) * eps;
}

// ---------------- kernel 7: beta offsets ----------------
__global__ void qg_offsets_kernel(const float* __restrict__ P,
                                  const int* __restrict__ rowpix,
                                  float* __restrict__ ooff, int T) {
    int j = blockIdx.x * blockDim.x + threadIdx.x;
    if (j >= T) return;
    int n = rowpix[j];
    float xa = P[(size_t)n * OUTC + 2 * QD + 0];
    float xb = P[(size_t)n * OUTC + 2 * QD + 1];
    float ya = P[(size_t)n * OUTC + 2 * QD + 2];
    float yb = P[(size_t)n * OUTC + 2 * QD + 3];
    float x = rbeta(xa, xb, 0x68E31DA4u ^ (unsigned)j);
    float y = rbeta(ya, yb, 0x1B56C4E9u ^ (unsigned)j);
    ooff[2 * j + 0] = x;
    ooff[2 * j + 1] = y;
}

// ---------------- launcher ----------------
extern "C" void kernel_launch(void* const* d_in, const int* in_sizes, int n_in,
                              void* d_out, int out_size, void* d_ws, size_t ws_size,
                              hipStream_t stream) {
    const int*   idx    = (const int*)  d_in[0];   // (N,3)
    const float* vals   = (const float*)d_in[1];   // (N,32)
    const float* logits = (const float*)d_in[2];   // (N,8)
    const float* w1     = (const float*)d_in[3];   // (3,3,32,64)
    const float* b1     = (const float*)d_in[4];   // (64)
    const float* w2     = (const float*)d_in[5];   // (1,1,64,68)
    const float* b2     = (const float*)d_in[6];   // (68)

    const int N = in_sizes[0] / 3;
    const int T = N * (NCLS - 1);

    // workspace layout
    const size_t DENSE_ELEMS = (size_t)BB * HH * WW * CIN;   // 16,777,216 floats
    float* dense  = (float*)d_ws;
    int*   counts = (int*)(dense + DENSE_ELEMS);
    int*   off    = counts + N;            // N+1
    int*   rowpix = off + (N + 1);         // T
    float* P      = (float*)(rowpix + T);  // N*68

    // output layout: queries (T*QD f32) | qidx (T*3 i32) | offsets (T*2 f32)
    float* qout     = (float*)d_out;
    int*   qidx_out = (int*)(qout + (size_t)T * QD);
    float* ooff     = (float*)(qidx_out + (size_t)T * 3);

    const int TPB = 256;

    // 0) zero dense
    {
        int n4 = (int)(DENSE_ELEMS / 4);
        qg_zero_kernel<<<(n4 + TPB - 1) / TPB, TPB, 0, stream>>>((float4*)dense, n4);
    }
    // 1) scatter
    qg_scatter_kernel<<<(N * CIN + TPB - 1) / TPB, TPB, 0, stream>>>(idx, vals, dense, N);
    // 2) counts
    qg_counts_kernel<<<(N + TPB - 1) / TPB, TPB, 0, stream>>>(logits, counts, N);
    // 3) scan
    qg_scan_kernel<<<1, 256, 0, stream>>>(counts, off, N);
    // 4) fill + expand (qidx + rowpix)
    qg_fill_kernel<<<(T + TPB - 1) / TPB, TPB, 0, stream>>>(idx, qidx_out, rowpix, N, T);
    qg_expand_kernel<<<(N + TPB - 1) / TPB, TPB, 0, stream>>>(idx, off, counts, qidx_out, rowpix, N, T);
    // 5) fused sparse conv via WMMA + async-to-LDS staging
    {
        int nBlocks = (N + CONV_PIX - 1) / CONV_PIX;
        qg_conv_wmma_kernel<<<nBlocks, CONV_WAVES * 32, 0, stream>>>(
            idx, w1, b1, w2, b2, dense, P, N);
    }
    // 6) queries
    qg_queries_kernel<<<(T * QD + TPB - 1) / TPB, TPB, 0, stream>>>(P, rowpix, qout, T);
    // 7) beta offsets
    qg_offsets_kernel<<<(T + TPB - 1) / TPB, TPB, 0, stream>>>(P, rowpix, ooff, T);
}